// RootMLP_45698452029677
// MI455X (gfx1250) — compile-verified
//
#include <hip/hip_runtime.h>
#include <hip/hip_bf16.h>

// Batched per-sample MLP (hypernetwork): dims 1 -> 128 x4 -> 64, silu hidden,
// head = [tanh(mean), exp(0.5*clip(logvar,-4,4))].
// HBM-bound (~951 MB weights, AI ~ 0.5 flop/byte) -> stream weights once,
// one wave32 per sample, fp32 matvecs via V_WMMA_F32_16X16X4_F32 with the
// activation vector broadcast across B's 16 columns.

typedef float v2f __attribute__((ext_vector_type(2)));
typedef float v8f __attribute__((ext_vector_type(8)));

#define MLP_WIDTH 128
#define MLP_OUT 64
#define MLP_TOTAL_PARAMS 58048
#define WAVES_PER_BLOCK 8

__device__ __forceinline__ float silu_f(float v) {
    return v / (1.0f + __expf(-v));
}

// One 128B cacheline per lane per step -> global_prefetch_b8 stream of the
// NEXT layer's weights while the current layer computes.
__device__ __forceinline__ void prefetch_span(const float* p, int nfloats, int lane) {
    for (int i = lane * 32; i < nfloats; i += 32 * 32)
        __builtin_prefetch(p + i, 0, 3);
}

// One fully-connected layer (in=128, out=MT*16) computed by one wave with
// V_WMMA_F32_16X16X4_F32.  xin/yout are this wave's LDS activation buffers.
// B fragment = activation broadcast across all 16 columns, so every column of
// D holds y; lanes 0 and 16 write the 16 rows of each M-tile back to LDS.
template<int MT, bool ACT>
__device__ __forceinline__ void layer_wmma(const float* __restrict__ W,
                                           const float* __restrict__ bias,
                                           const float* xin, float* yout,
                                           int lane) {
    const int half = lane >> 4;   // 0: lanes 0-15 (K=0,1), 1: lanes 16-31 (K=2,3)
    const int row  = lane & 15;

    // Hoist all B fragments (the activation vector) out of the M loop.
    v2f bfrag[32];
    #pragma unroll
    for (int kk = 0; kk < 32; ++kk)
        bfrag[kk] = *(const v2f*)(xin + kk * 4 + 2 * half);   // ds_load_b64

    #pragma unroll
    for (int m = 0; m < MT; ++m) {
        const float* wrow = W + (size_t)(m * 16 + row) * MLP_WIDTH + 2 * half;
        v8f acc0 = {0.f, 0.f, 0.f, 0.f, 0.f, 0.f, 0.f, 0.f};
        v8f acc1 = {0.f, 0.f, 0.f, 0.f, 0.f, 0.f, 0.f, 0.f};
        #pragma unroll
        for (int kk = 0; kk < 32; kk += 2) {
            v2f a0 = *(const v2f*)(wrow + kk * 4);       // global_load_b64
            v2f a1 = *(const v2f*)(wrow + kk * 4 + 4);
            acc0 = __builtin_amdgcn_wmma_f32_16x16x4_f32(
                false, a0, false, bfrag[kk],     (short)0, acc0, false, false);
            acc1 = __builtin_amdgcn_wmma_f32_16x16x4_f32(
                false, a1, false, bfrag[kk + 1], (short)0, acc1, false, false);
        }
        v8f acc = acc0 + acc1;
        // D layout (16x16 f32): VGPR r = row m*16 + half*8 + r, all 16 cols equal.
        if (row == 0) {
            #pragma unroll
            for (int r = 0; r < 8; ++r)
                yout[m * 16 + half * 8 + r] = acc[r];
        }
    }
    __syncthreads();

    // bias + activation, fully coalesced
    #pragma unroll
    for (int o = 0; o < MT * 16; o += 32) {
        float v = yout[o + lane] + bias[o + lane];
        if (ACT) v = silu_f(v);
        yout[o + lane] = v;
    }
    __syncthreads();
}

__global__ __launch_bounds__(WAVES_PER_BLOCK * 32)
void RootMLP_45698452029677_kernel(const float* __restrict__ x,
                                   const float* __restrict__ w,
                                   float* __restrict__ out,
                                   int batch) {
    __shared__ float sact[WAVES_PER_BLOCK][2][MLP_WIDTH];
    const int lane = threadIdx.x & 31;
    const int wave = threadIdx.x >> 5;
    const int b    = blockIdx.x * WAVES_PER_BLOCK + wave;
    // Clamp instead of early-exit: keeps barriers uniform and EXEC all-ones
    // for every WMMA (redundant tail waves recompute sample batch-1).
    const int bb = (b < batch) ? b : (batch - 1);

    const float* wp = w + (size_t)bb * MLP_TOTAL_PARAMS;
    float* bufA = sact[wave][0];
    float* bufB = sact[wave][1];

    // Flat parameter offsets (floats), row-major W then bias per layer.
    const float* W0 = wp;           const float* B0 = wp + 128;
    const float* W1 = wp + 256;     const float* B1 = wp + 16640;
    const float* W2 = wp + 16768;   const float* B2 = wp + 33152;
    const float* W3 = wp + 33280;   const float* B3 = wp + 49664;
    const float* W4 = wp + 49792;   const float* B4 = wp + 57984;

    prefetch_span(W1, 16512, lane);

    // Layer 0: 1 -> 128, scalar broadcast input.
    const float xv = x[bb];
    for (int o = lane; o < MLP_WIDTH; o += 32)
        bufA[o] = silu_f(W0[o] * xv + B0[o]);
    __syncthreads();

    prefetch_span(W2, 16512, lane);
    layer_wmma<8, true >(W1, B1, bufA, bufB, lane);
    prefetch_span(W3, 16512, lane);
    layer_wmma<8, true >(W2, B2, bufB, bufA, lane);
    prefetch_span(W4, 8256, lane);
    layer_wmma<8, true >(W3, B3, bufA, bufB, lane);
    layer_wmma<4, false>(W4, B4, bufB, bufA, lane);

    // Head: mean = tanh(y[0:32]); std = exp(0.5 * clip(y[32:64], -4, 4)).
    const float mval = tanhf(bufA[lane]);
    float lv = bufA[32 + lane];
    lv = fminf(4.0f, fmaxf(-4.0f, lv));
    const float sval = __expf(0.5f * lv);
    if (b < batch) {
        out[(size_t)b * MLP_OUT + lane]      = mval;
        out[(size_t)b * MLP_OUT + 32 + lane] = sval;
    }
}

extern "C" void kernel_launch(void* const* d_in, const int* in_sizes, int n_in,
                              void* d_out, int out_size, void* d_ws, size_t ws_size,
                              hipStream_t stream) {
    const float* x  = (const float*)d_in[0];   // (BATCH, 1) fp32
    const float* fw = (const float*)d_in[1];   // (BATCH, 58048) fp32
    float* out = (float*)d_out;                // (BATCH, 64) fp32
    const int batch = in_sizes[0];             // IN_DIM == 1
    const int blocks = (batch + WAVES_PER_BLOCK - 1) / WAVES_PER_BLOCK;
    RootMLP_45698452029677_kernel<<<blocks, WAVES_PER_BLOCK * 32, 0, stream>>>(
        x, fw, out, batch);
}